// GCN_10170482556987
// MI455X (gfx1250) — compile-verified
//
#include <hip/hip_runtime.h>

#define N_NODES 50000
#define N_EDGES 800000
#define IN_CH 512
#define HID 128
#define OUT_CH 2
#define N_GRAPHS 64
#define EPS_BN 1e-5f

typedef __attribute__((ext_vector_type(2))) float v2f;
typedef __attribute__((ext_vector_type(8))) float v8f;
typedef int v4i __attribute__((vector_size(16)));  // b128 payload type

// CDNA5 async global->LDS copy path (ASYNCcnt-tracked), guarded so the file
// still compiles if this toolchain doesn't declare the builtins.
#if defined(__has_builtin)
#if __has_builtin(__builtin_amdgcn_global_load_async_to_lds_b128) && \
    __has_builtin(__builtin_amdgcn_s_wait_asynccnt)
#define USE_ASYNC_LDS 1
#endif
#endif

// ---------------------------------------------------------------------------
// Transpose weights once: Wt[n][k] = W[k][n], so the WMMA B fragment
// {W[k][n], W[k+1][n]} becomes one aligned 8-byte load.
// ---------------------------------------------------------------------------
template <int K>
__global__ void transpose_w(const float* __restrict__ W, float* __restrict__ Wt) {
  const int t = blockIdx.x * 256 + threadIdx.x;  // over K*HID
  if (t >= K * HID) return;
  const int k = t >> 7;          // t / HID (HID=128)
  const int n = t & (HID - 1);
  Wt[n * K + k] = W[t];
}

// ---------------------------------------------------------------------------
// WMMA GEMM: C[M x 128] = A[M x K] @ W[K x 128]   (Wt is W transposed)
// M multiple of 16 (50000 = 3125*16). Workgroup = 256 threads = 8 wave32;
// block owns a 16-row M tile staged in LDS; wave w owns columns [16w,16w+16).
// V_WMMA_F32_16X16X4_F32 fragments:
//   A (16x4): v0 = A[m][k0+2*half], v1 = A[m][k0+2*half+1]   (m = lane&15)
//   B (4x16): v0 = W[k0+2*half][n], v1 = W[k0+2*half+1][n]   (n = lane&15)
//   C/D (16x16): vgpr r -> row r + 8*half, col = lane&15
// ---------------------------------------------------------------------------
template <int K>
__global__ __launch_bounds__(256) void gcn_gemm_wmma(
    const float* __restrict__ A, const float* __restrict__ Wt,
    float* __restrict__ C) {
  __shared__ float Atile[16 * K];
  const int m0 = blockIdx.x * 16;
  const float* __restrict__ Ag = A + (size_t)m0 * K;  // 16*K contiguous floats

  constexpr int NV = (16 * K) / 4;  // float4 chunks
#if USE_ASYNC_LDS
  for (int t = threadIdx.x; t < NV; t += 256) {
    __builtin_amdgcn_global_load_async_to_lds_b128(
        (__attribute__((address_space(1))) v4i*)(Ag + 4 * t),
        (__attribute__((address_space(3))) v4i*)(Atile + 4 * t),
        0, 0);
  }
  __builtin_amdgcn_s_wait_asynccnt(0);
#else
  for (int t = threadIdx.x; t < NV; t += 256) {
    ((float4*)Atile)[t] = ((const float4*)Ag)[t];
  }
#endif
  __syncthreads();

  const int wave = threadIdx.x >> 5;
  const int lane = threadIdx.x & 31;
  const int half = lane >> 4;   // 0: lanes 0-15, 1: lanes 16-31
  const int mrow = lane & 15;
  const int ncol = lane & 15;
  const int n0   = wave * 16;

  const float* __restrict__ ash  = Atile + mrow * K + 2 * half;
  const float* __restrict__ wrow = Wt + (size_t)(n0 + ncol) * K + 2 * half;

  v8f acc = {};
#pragma unroll 8
  for (int k0 = 0; k0 < K; k0 += 4) {
    v2f a = *(const v2f*)(ash + k0);   // ds_load_b64, immediate offset
    v2f b = *(const v2f*)(wrow + k0);  // global_load_b64, immediate offset
    acc = __builtin_amdgcn_wmma_f32_16x16x4_f32(
        false, a, false, b, (short)0, acc, false, false);
  }

#pragma unroll
  for (int r = 0; r < 8; ++r) {
    C[(size_t)(m0 + r + 8 * half) * HID + n0 + ncol] = acc[r];
  }
}

// ---------------------------------------------------------------------------
// Degree (with self loop) and D^-1/2
// ---------------------------------------------------------------------------
__global__ void deg_init(float* __restrict__ deg) {
  const int i = blockIdx.x * 256 + threadIdx.x;
  if (i < N_NODES) deg[i] = 1.0f;  // self loop
}

__global__ void deg_edges(const int* __restrict__ dst, float* __restrict__ deg) {
  const int e = blockIdx.x * 256 + threadIdx.x;
  if (e < N_EDGES) atomicAdd(&deg[dst[e]], 1.0f);
}

__global__ void deg_rsqrt(const float* __restrict__ deg, float* __restrict__ dinv) {
  const int i = blockIdx.x * 256 + threadIdx.x;
  if (i < N_NODES) dinv[i] = rsqrtf(deg[i]);
}

// ---------------------------------------------------------------------------
// Aggregation init: out[i] = dinv[i]^2 * h[i]  (self-loop term)
// ---------------------------------------------------------------------------
__global__ void agg_self(const float* __restrict__ h,
                         const float* __restrict__ dinv,
                         float* __restrict__ out) {
  const int t = blockIdx.x * 256 + threadIdx.x;  // over N_NODES * 32 (float4)
  if (t >= N_NODES * (HID / 4)) return;
  const int i = t >> 5;  // t / 32
  float s = dinv[i];
  s = s * s;
  const float4 v = ((const float4*)h)[t];
  float4 o;
  o.x = v.x * s; o.y = v.y * s; o.z = v.z * s; o.w = v.w * s;
  ((float4*)out)[t] = o;
}

// One wave32 per edge; each lane handles 4 consecutive channels (float4).
__global__ __launch_bounds__(256) void agg_edges(
    const int* __restrict__ src, const int* __restrict__ dst,
    const float* __restrict__ h, const float* __restrict__ dinv,
    float* __restrict__ out) {
  const int wid = (blockIdx.x * 256 + threadIdx.x) >> 5;  // edge id
  if (wid >= N_EDGES) return;
  const int lane = threadIdx.x & 31;
  const int s = src[wid];
  const int d = dst[wid];
  const float nrm = dinv[s] * dinv[d];
  const float4 v = ((const float4*)(h + (size_t)s * HID))[lane];
  float* op = out + (size_t)d * HID + lane * 4;
  atomicAdd(op + 0, v.x * nrm);
  atomicAdd(op + 1, v.y * nrm);
  atomicAdd(op + 2, v.z * nrm);
  atomicAdd(op + 3, v.w * nrm);
}

// ---------------------------------------------------------------------------
// Fused bias + BatchNorm(eval) + ReLU, float4 vectorized
// ---------------------------------------------------------------------------
__global__ void bn_relu4(const float4* __restrict__ agg,
                         const float4* __restrict__ b,
                         const float4* __restrict__ gamma,
                         const float4* __restrict__ beta,
                         const float4* __restrict__ rm,
                         const float4* __restrict__ rv,
                         float4* __restrict__ out) {
  const int t = blockIdx.x * 256 + threadIdx.x;  // over N_NODES * 32
  if (t >= N_NODES * (HID / 4)) return;
  const int c = t & 31;  // float4 channel group
  const float4 v = agg[t], bb = b[c], g = gamma[c], be = beta[c],
               m = rm[c], va = rv[c];
  float4 o;
  o.x = fmaxf((v.x + bb.x - m.x) * rsqrtf(va.x + EPS_BN) * g.x + be.x, 0.0f);
  o.y = fmaxf((v.y + bb.y - m.y) * rsqrtf(va.y + EPS_BN) * g.y + be.y, 0.0f);
  o.z = fmaxf((v.z + bb.z - m.z) * rsqrtf(va.z + EPS_BN) * g.z + be.z, 0.0f);
  o.w = fmaxf((v.w + bb.w - m.w) * rsqrtf(va.w + EPS_BN) * g.w + be.w, 0.0f);
  out[t] = o;
}

// ---------------------------------------------------------------------------
// Global mean pool + linear head
// ---------------------------------------------------------------------------
__global__ void pool_zero(float* __restrict__ sums, float* __restrict__ counts) {
  const int t = blockIdx.x * 256 + threadIdx.x;
  if (t < N_GRAPHS * HID) sums[t] = 0.0f;
  if (t < N_GRAPHS) counts[t] = 0.0f;
}

__global__ __launch_bounds__(256) void pool_acc(
    const float* __restrict__ h, const int* __restrict__ batch,
    float* __restrict__ sums, float* __restrict__ counts) {
  const int wid = (blockIdx.x * 256 + threadIdx.x) >> 5;  // node id
  if (wid >= N_NODES) return;
  const int lane = threadIdx.x & 31;
  const int g = batch[wid];
  const float4 v = ((const float4*)(h + (size_t)wid * HID))[lane];
  float* sp = sums + g * HID + lane * 4;
  atomicAdd(sp + 0, v.x);
  atomicAdd(sp + 1, v.y);
  atomicAdd(sp + 2, v.z);
  atomicAdd(sp + 3, v.w);
  if (lane == 0) atomicAdd(&counts[g], 1.0f);
}

__global__ void head(const float* __restrict__ sums,
                     const float* __restrict__ counts,
                     const float* __restrict__ Wl,
                     const float* __restrict__ bl,
                     float* __restrict__ out) {
  const int t = threadIdx.x;  // 128 = 64 graphs * 2 outputs
  if (t >= N_GRAPHS * OUT_CH) return;
  const int g = t / OUT_CH;
  const int o = t - g * OUT_CH;
  const float inv = 1.0f / fmaxf(counts[g], 1.0f);
  float acc = bl[o];
  for (int k = 0; k < HID; ++k) {
    acc += sums[g * HID + k] * inv * Wl[k * OUT_CH + o];
  }
  out[t] = acc;
}

// ---------------------------------------------------------------------------
extern "C" void kernel_launch(void* const* d_in, const int* in_sizes, int n_in,
                              void* d_out, int out_size, void* d_ws, size_t ws_size,
                              hipStream_t stream) {
  const float* x      = (const float*)d_in[0];
  const int*   ei     = (const int*)d_in[1];
  const int*   batch  = (const int*)d_in[2];
  const float* W1     = (const float*)d_in[3];
  const float* b1     = (const float*)d_in[4];
  const float* gamma1 = (const float*)d_in[5];
  const float* beta1  = (const float*)d_in[6];
  const float* rm1    = (const float*)d_in[7];
  const float* rv1    = (const float*)d_in[8];
  const float* W2     = (const float*)d_in[9];
  const float* b2     = (const float*)d_in[10];
  const float* gamma2 = (const float*)d_in[11];
  const float* beta2  = (const float*)d_in[12];
  const float* rm2    = (const float*)d_in[13];
  const float* rv2    = (const float*)d_in[14];
  const float* Wl     = (const float*)d_in[15];
  const float* bl     = (const float*)d_in[16];

  const int* src = ei;            // edge_index[0]
  const int* dst = ei + N_EDGES;  // edge_index[1]

  // Workspace layout (~52 MB)
  float* bufA   = (float*)d_ws;              // [N_NODES * HID]
  float* bufB   = bufA + N_NODES * HID;      // [N_NODES * HID]
  float* deg    = bufB + N_NODES * HID;      // [N_NODES]
  float* dinv   = deg + N_NODES;             // [N_NODES]
  float* sums   = dinv + N_NODES;            // [N_GRAPHS * HID]
  float* counts = sums + N_GRAPHS * HID;     // [N_GRAPHS]
  float* wt1    = counts + N_GRAPHS;         // [HID * IN_CH]
  float* wt2    = wt1 + HID * IN_CH;         // [HID * HID]

  const dim3 B(256);
  const int gNodes = (N_NODES + 255) / 256;
  const int gEdges = (N_EDGES + 255) / 256;
  const int gNode4 = (N_NODES * (HID / 4) + 255) / 256;  // float4 elementwise
  const int gEdgeW = (N_EDGES * 32 + 255) / 256;         // wave per edge
  const int gNodeW = (N_NODES * 32 + 255) / 256;         // wave per node

  // weight transposes (once per call; tiny)
  transpose_w<IN_CH><<<(IN_CH * HID + 255) / 256, B, 0, stream>>>(W1, wt1);
  transpose_w<HID><<<(HID * HID + 255) / 256, B, 0, stream>>>(W2, wt2);

  // degrees / normalization (shared by both layers)
  deg_init<<<gNodes, B, 0, stream>>>(deg);
  deg_edges<<<gEdges, B, 0, stream>>>(dst, deg);
  deg_rsqrt<<<gNodes, B, 0, stream>>>(deg, dinv);

  // ---- layer 1 ----
  gcn_gemm_wmma<IN_CH><<<N_NODES / 16, B, 0, stream>>>(x, wt1, bufA);
  agg_self<<<gNode4, B, 0, stream>>>(bufA, dinv, bufB);
  agg_edges<<<gEdgeW, B, 0, stream>>>(src, dst, bufA, dinv, bufB);
  bn_relu4<<<gNode4, B, 0, stream>>>((const float4*)bufB, (const float4*)b1,
                                     (const float4*)gamma1, (const float4*)beta1,
                                     (const float4*)rm1, (const float4*)rv1,
                                     (float4*)bufA);

  // ---- layer 2 ----
  gcn_gemm_wmma<HID><<<N_NODES / 16, B, 0, stream>>>(bufA, wt2, bufB);
  agg_self<<<gNode4, B, 0, stream>>>(bufB, dinv, bufA);
  agg_edges<<<gEdgeW, B, 0, stream>>>(src, dst, bufB, dinv, bufA);
  bn_relu4<<<gNode4, B, 0, stream>>>((const float4*)bufA, (const float4*)b2,
                                     (const float4*)gamma2, (const float4*)beta2,
                                     (const float4*)rm2, (const float4*)rv2,
                                     (float4*)bufB);

  // ---- pool + head ----
  pool_zero<<<(N_GRAPHS * HID + 255) / 256, B, 0, stream>>>(sums, counts);
  pool_acc<<<gNodeW, B, 0, stream>>>(bufB, batch, sums, counts);
  head<<<1, 128, 0, stream>>>(sums, counts, Wl, bl, (float*)d_out);
}